// TreecrfLossSRL_44659069943944
// MI455X (gfx1250) — compile-verified
//
#include <hip/hip_runtime.h>

typedef __attribute__((ext_vector_type(2))) float v2f;
typedef __attribute__((ext_vector_type(8))) float v8f;

#define NN      256
#define STR     257            // chart row stride (odd -> conflict-free columns)
#define LOG2E   1.4426950408889634f
#define LN2     0.6931471805599453f
#define NEGI    (-1.0e30f)

// single-instruction TRANS ops on gfx1250
#define EXP2(x) __builtin_amdgcn_exp2f(x)   // v_exp_f32: 2^x
#define LOG2(x) __builtin_amdgcn_logf(x)    // v_log_f32: log2(x)

// streaming logsumexp (base 2) on a (max, scaled-sum) pair
__device__ __forceinline__ void lse_upd(float& M, float& S, float t) {
    float Mn = fmaxf(M, t);
    S = S * EXP2(M - Mn) + EXP2(t - Mn);
    M = Mn;
}

// D score (base 2) for inclusive span [i, j] of batch b; cs=0 -> allv, cs=1 -> ob
__device__ __forceinline__ float dscore2(const float* __restrict__ logits,
                                         const int* __restrict__ ind,
                                         int b, int cs, int i, int j) {
    size_t base = ((size_t)b * NN + i) * NN + j;
    const float2* lp = (const float2*)logits;
    float2 l = lp[base];
    if (cs == 0) {
        // logaddexp(l0, l1) computed directly in base 2
        float mx = fmaxf(l.x, l.y), mn = fminf(l.x, l.y);
        return mx * LOG2E + LOG2(1.0f + EXP2((mn - mx) * LOG2E));
    } else {
        float d = (ind[base] == 2) ? l.y : l.x;     // NEG branches collapse exactly
        return d * LOG2E;
    }
}

__global__ __launch_bounds__(256)
void TreecrfCykBlocked_kernel(const float* __restrict__ logits,
                              const int*   __restrict__ ind,
                              float*       __restrict__ pair_out) {
    __shared__ float chart[NN * STR];        // 263,168 B: inside chart, base-2
    __shared__ float accM[8][16][16];        // per-wave GEMM accumulator (max)
    __shared__ float accS[8][16][16];        // per-wave GEMM accumulator (scaled sum)
    __shared__ float scl[8][32];             // per-wave row/col maxes (mL | mR)

    const int b  = blockIdx.x >> 1;
    const int cs = blockIdx.x & 1;
    const int lane = threadIdx.x & 31;
    const int wv   = __builtin_amdgcn_readfirstlane((int)(threadIdx.x >> 5));
    const int m16  = lane & 15;
    const int half = lane >> 4;

    // warm L2/L0 for this batch's logits (gfx1250 global_prefetch path)
    __builtin_prefetch((const void*)(logits + (size_t)b * NN * NN * 2 + threadIdx.x * 32), 0, 0);

    // ---------------- d = 0 : diagonal 16x16 blocks, scalar micro-CYK ----------------
    for (int Iblk = wv; Iblk < 16; Iblk += 8) {
        int i0 = Iblk * 16;
        if (lane < 16)
            chart[(i0 + lane) * STR + (i0 + lane)] = dscore2(logits, ind, b, cs, i0 + lane, i0 + lane);
        __builtin_amdgcn_wave_barrier();
        for (int w = 1; w < 16; ++w) {
            bool act = (lane < 16 - w);
            int i = i0 + lane, j = i + w;
            float val = 0.f;
            if (act) {
                float M = NEGI, S = 0.f;
                #pragma unroll 1
                for (int k = i; k < j; ++k)
                    lse_upd(M, S, chart[i * STR + k] + chart[(k + 1) * STR + j]);
                val = dscore2(logits, ind, b, cs, i, j) + M + LOG2(S);
            }
            __builtin_amdgcn_wave_barrier();
            if (act) chart[i * STR + j] = val;
            __builtin_amdgcn_wave_barrier();
        }
    }
    __syncthreads();

    // ---------------- d = 1..15 : block anti-diagonals ----------------
    for (int d = 1; d < 16; ++d) {
        for (int Iblk = wv; Iblk < 16 - d; Iblk += 8) {
            const int Jblk = Iblk + d;
            const int i0 = Iblk * 16, j0 = Jblk * 16;

            float Mo[8], So[8];
            #pragma unroll
            for (int r = 0; r < 8; ++r) { Mo[r] = NEGI; So[r] = 0.f; }

            // ---- middle k-blocks: exp2-scaled fp32 WMMA GEMMs ----
            #pragma unroll 1
            for (int K = Iblk + 1; K < Jblk; ++K) {
                const int k0 = K * 16;
                // lanes 0-15: row maxes of left slice; lanes 16-31: col maxes of right slice
                float mv = NEGI;
                if (half == 0) {
                    int i = i0 + m16;
                    #pragma unroll
                    for (int k = 0; k < 16; ++k) mv = fmaxf(mv, chart[i * STR + k0 + k]);
                } else {
                    int j = j0 + m16;
                    #pragma unroll
                    for (int k = 0; k < 16; ++k) mv = fmaxf(mv, chart[(k0 + k + 1) * STR + j]);
                }
                scl[wv][lane] = mv;
                __builtin_amdgcn_wave_barrier();
                const float mL = scl[wv][m16];          // A row = lane%16
                const float mR = scl[wv][16 + m16];     // B col = lane%16

                v8f S4 = {0.f, 0.f, 0.f, 0.f, 0.f, 0.f, 0.f, 0.f};
                #pragma unroll
                for (int kk = 0; kk < 4; ++kk) {
                    int ka = k0 + kk * 4 + half * 2;    // ISA 16x4 f32 fragment layout
                    int ar = i0 + m16;                  // A row
                    int bc = j0 + m16;                  // B col
                    v2f a, bb;
                    a.x  = EXP2(chart[ar * STR + ka]           - mL);
                    a.y  = EXP2(chart[ar * STR + ka + 1]       - mL);
                    bb.x = EXP2(chart[(ka + 1) * STR + bc]     - mR);
                    bb.y = EXP2(chart[(ka + 2) * STR + bc]     - mR);
                    S4 = __builtin_amdgcn_wmma_f32_16x16x4_f32(
                            false, a, false, bb, (short)0, S4, false, false);
                }
                // fold this k-block into per-output (max, scaled-sum)
                #pragma unroll
                for (int r = 0; r < 8; ++r) {
                    int m = r + 8 * half;
                    float off = scl[wv][m] + scl[wv][16 + m16];   // mL[m] + mR[n]
                    float Mn = fmaxf(Mo[r], off);
                    So[r] = So[r] * EXP2(Mo[r] - Mn) + S4[r] * EXP2(off - Mn);
                    Mo[r] = Mn;
                }
            }

            // dump distributed accumulators to per-wave LDS tiles
            #pragma unroll
            for (int r = 0; r < 8; ++r) {
                int m = r + 8 * half;
                accM[wv][m][m16] = Mo[r];
                accS[wv][m][m16] = So[r];
            }
            __builtin_amdgcn_wave_barrier();

            // ---- fringe splits (k in corner diagonal blocks), intra-block wavefront ----
            const int wlo = j0 - (i0 + 15), whi = (j0 + 15) - i0;
            #pragma unroll 1
            for (int w = wlo; w <= whi; ++w) {
                int i = i0 + lane, j = i + w;
                bool act = (lane < 16) && (j >= j0) && (j <= j0 + 15);
                float val = 0.f;
                if (act) {
                    float M = accM[wv][i - i0][j - j0];
                    float S = accS[wv][i - i0][j - j0];
                    #pragma unroll 1
                    for (int k = i; k <= i0 + 15; ++k)        // k in block I
                        lse_upd(M, S, chart[i * STR + k] + chart[(k + 1) * STR + j]);
                    #pragma unroll 1
                    for (int k = j0; k < j; ++k)              // k in block J
                        lse_upd(M, S, chart[i * STR + k] + chart[(k + 1) * STR + j]);
                    val = dscore2(logits, ind, b, cs, i, j) + M + LOG2(S);
                }
                __builtin_amdgcn_wave_barrier();
                if (act) chart[i * STR + j] = val;
                __builtin_amdgcn_wave_barrier();
            }
        }
        __syncthreads();
    }

    if (threadIdx.x == 0)
        pair_out[blockIdx.x] = chart[0 * STR + (NN - 1)] * LN2;   // logZ in nats
}

__global__ void TreecrfReduce_kernel(const float* __restrict__ pr, float* __restrict__ out) {
    __shared__ float sm[64];
    int t = threadIdx.x;
    sm[t] = pr[2 * t] - pr[2 * t + 1];     // logz(b) - marginals(b)
    __syncthreads();
    #pragma unroll
    for (int s = 32; s > 0; s >>= 1) {
        if (t < s) sm[t] += sm[t + s];
        __syncthreads();
    }
    if (t == 0) out[0] = sm[0] / 16384.0f;
}

extern "C" void kernel_launch(void* const* d_in, const int* in_sizes, int n_in,
                              void* d_out, int out_size, void* d_ws, size_t ws_size,
                              hipStream_t stream) {
    (void)in_sizes; (void)n_in; (void)out_size; (void)ws_size;
    const float* logits = (const float*)d_in[0];   // [64,256,256,2] f32
    const int*   ind    = (const int*)d_in[1];     // [64,256,256] i32
    float* pair = (float*)d_ws;                    // 128 floats of scratch

    TreecrfCykBlocked_kernel<<<128, 256, 0, stream>>>(logits, ind, pair);
    TreecrfReduce_kernel<<<1, 64, 0, stream>>>(pair, (float*)d_out);
}